// MultiViewBoxFusion_274877907269
// MI455X (gfx1250) — compile-verified
//
#include <hip/hip_runtime.h>
#include <stdint.h>

// MultiViewBoxFusion for MI455X (gfx1250, wave32).
// Pipeline: bitonic argsort -> AABB prep -> TDM-staged 8192x8192 IoU bitmask
// (wave32 ballot) -> blocked serial NMS -> per-pair fusion.
// Workspace use: M*4 (order) + M*32 (sorted boxes) + M*32 (prep SoA)
//              + M*(M/32)*4 (8MB bitmask) + (M/32)*4 (keep words) ~= 8.93 MB.

#define THRF 0.3f
#define EPSF 1e-6f

typedef unsigned int u32;
typedef unsigned long long u64;
typedef u32 uv4 __attribute__((ext_vector_type(4)));
typedef int iv4 __attribute__((ext_vector_type(4)));
typedef int iv8 __attribute__((ext_vector_type(8)));

#define MCAP 8192  // 2*N for N=4096 (power of two, required by bitonic sort)

// ---------------------------------------------------------------------------
// Kernel 1: single-workgroup bitonic argsort of -scores (stable via idx key).
// Key = (~totalOrder(score)) << 32 | idx : ascending u64 sort == descending
// score with ascending-index tiebreak (== stable argsort of -scores).
// ---------------------------------------------------------------------------
__global__ __launch_bounds__(1024) void sort_kernel(const float* __restrict__ s0,
                                                    const float* __restrict__ s1,
                                                    int N, int M, int* __restrict__ order) {
  __shared__ u64 keys[MCAP];  // 64 KB LDS
  const int tid = threadIdx.x;
  for (int i = tid; i < M; i += blockDim.x) {
    float sc = (i < N) ? s0[i] : s1[i - N];
    u32 b = __float_as_uint(sc);
    b = (b & 0x80000000u) ? ~b : (b | 0x80000000u);  // monotonic total order
    keys[i] = ((u64)(~b) << 32) | (u32)i;
  }
  __syncthreads();
  for (int k = 2; k <= M; k <<= 1) {
    for (int j = k >> 1; j > 0; j >>= 1) {
      for (int i = tid; i < M; i += blockDim.x) {
        int p = i ^ j;
        if (p > i) {
          bool up = ((i & k) == 0);
          u64 a = keys[i], b2 = keys[p];
          if ((a > b2) == up) { keys[i] = b2; keys[p] = a; }
        }
      }
      __syncthreads();
    }
  }
  for (int i = tid; i < M; i += blockDim.x)
    order[i] = (int)(u32)(keys[i] & 0xFFFFFFFFu);
}

// ---------------------------------------------------------------------------
// Kernel 2: per-sorted-box preprocessing.
// Rotated AABB extent: gx = lx*c - lz*s + cx over corners => ext = |dx c|+|dz s|.
// prep is SoA [8][M]: ymin, ymax, h, x1, x2, z1, z2, area (TDM-friendly, 256 KB).
// sboxes is AoS [M][8]: the sorted 7-float box (padded).
// ---------------------------------------------------------------------------
__global__ void prep_kernel(const float* __restrict__ b0, const float* __restrict__ b1,
                            const int* __restrict__ order, int N, int M,
                            float* __restrict__ prep, float* __restrict__ sboxes) {
  int p = blockIdx.x * blockDim.x + threadIdx.x;
  if (p >= M) return;
  int idx = order[p];
  const float* bx = (idx < N) ? (b0 + (size_t)idx * 7) : (b1 + (size_t)(idx - N) * 7);
  float cx = bx[0], cy = bx[1], cz = bx[2];
  float dx = bx[3] * 0.5f, dz = bx[4] * 0.5f, h = bx[5], yaw = bx[6];
  float co = cosf(yaw), si = sinf(yaw);
  float ex = fabsf(dx * co) + fabsf(dz * si);
  float ez = fabsf(dx * si) + fabsf(dz * co);
  float x1 = cx - ex, x2 = cx + ex, z1 = cz - ez, z2 = cz + ez;
  prep[0 * M + p] = cy - h * 0.5f;
  prep[1 * M + p] = cy + h * 0.5f;
  prep[2 * M + p] = h;
  prep[3 * M + p] = x1;
  prep[4 * M + p] = x2;
  prep[5 * M + p] = z1;
  prep[6 * M + p] = z2;
  prep[7 * M + p] = (x2 - x1) * (z2 - z1);
  float* sb = sboxes + (size_t)p * 8;
#pragma unroll
  for (int c = 0; c < 7; ++c) sb[c] = bx[c];
  sb[7] = 0.0f;
}

// ---------------------------------------------------------------------------
// Kernel 3: suppression bitmask. 256 threads (8 waves), 64 rows per block.
// Entire 256 KB prep table staged into LDS once per block via the Tensor Data
// Mover (one TDM descriptor: 2-D tensor 8 rows x M fp32, tile == tensor),
// then each wave computes 8 rows; 32 column-predicates are packed per word
// with wave32 ballot. Predicate is the divide-free equivalent of
//   inter/(union+eps) * yov/(hun+eps) >= 0.3
// ---------------------------------------------------------------------------
__global__ __launch_bounds__(256) void iou_mask_kernel(const float* __restrict__ prep,
                                                       u32* __restrict__ mask, int M) {
  extern __shared__ float lds_f[];  // 8*M floats = 256 KB dynamic LDS
  const int tid = threadIdx.x;
  const int lane = tid & 31;
  const int wave = tid >> 5;
  const int nw = M >> 5;

  if (wave == 0) {
    // ---- Tensor DMA descriptor (D#) per CDNA5 ISA ----
    u64 ga = (u64)(uintptr_t)prep;
    u32 ldsoff = __builtin_amdgcn_groupstaticsize();  // dynamic-LDS base offset
    uv4 g0;
    g0[0] = 0x1u;                                           // count=1, user mode
    g0[1] = ldsoff;                                         // lds_addr (bytes)
    g0[2] = (u32)ga;                                        // global_addr[31:0]
    g0[3] = (u32)((ga >> 32) & 0x01FFFFFFu) | (2u << 30);   // global_addr hi | type=2
    iv8 g1;
    g1[0] = (int)(2u << 16);                                // data_size = 4 B
    g1[1] = (int)(((u32)M & 0xFFFFu) << 16);                // tensor_dim0 lo16
    g1[2] = (int)((((u32)M >> 16) & 0xFFFFu) | (8u << 16)); // dim0 hi16 | tensor_dim1=8
    g1[3] = (int)(((u32)M & 0xFFFFu) << 16);                // dim1 hi16=0 | tile_dim0=M
    g1[4] = (int)8u;                                        // tile_dim1=8, tile_dim2=0
    g1[5] = (int)(u32)M;                                    // tensor_dim0_stride lo32
    g1[6] = 0;                                              // stride hi | dim1_stride lo
    g1[7] = 0;
    iv4 z4 = {0, 0, 0, 0};                                  // 2-D: groups 2/3 unused
    iv8 z8 = {0, 0, 0, 0, 0, 0, 0, 0};
    __builtin_amdgcn_tensor_load_to_lds(g0, g1, z4, z4, z8, 0);  // 6-arg form
    __builtin_amdgcn_s_wait_tensorcnt(0);
  }
  __syncthreads();

  const int base = blockIdx.x * 64;
  float R[8][8];  // 8 rows per wave, 8 features each (LDS broadcast reads)
#pragma unroll
  for (int t = 0; t < 8; ++t) {
    int r = base + wave * 8 + t;
#pragma unroll
    for (int c = 0; c < 8; ++c) R[t][c] = lds_f[c * M + r];
  }

  for (int cw = 0; cw < nw; ++cw) {
    int j = (cw << 5) + lane;
    float cy1 = lds_f[0 * M + j], cy2 = lds_f[1 * M + j];
    float chh = lds_f[2 * M + j], cx1 = lds_f[3 * M + j];
    float cx2 = lds_f[4 * M + j], cz1 = lds_f[5 * M + j];
    float cz2 = lds_f[6 * M + j], car = lds_f[7 * M + j];
#pragma unroll
    for (int t = 0; t < 8; ++t) {
      float yov = fminf(R[t][1], cy2) - fmaxf(R[t][0], cy1);
      yov = fmaxf(yov, 0.0f);
      float hun = R[t][2] + chh - yov;
      float xi = fminf(R[t][4], cx2) - fmaxf(R[t][3], cx1);
      xi = fmaxf(xi, 0.0f);
      float zi = fminf(R[t][6], cz2) - fmaxf(R[t][5], cz1);
      zi = fmaxf(zi, 0.0f);
      float inter = xi * zi;
      float uni = R[t][7] + car - inter;
      bool pred = inter * yov >= THRF * (uni + EPSF) * (hun + EPSF);
      u32 bits = __builtin_amdgcn_ballot_w32(pred);
      if (lane == t) mask[(size_t)(base + wave * 8 + t) * nw + cw] = bits;
    }
  }
}

// ---------------------------------------------------------------------------
// Kernel 4: blocked sequential NMS (single workgroup — the dependency chain
// is inherently serial in i). 32-row blocks: stage 32 KB of mask rows into
// LDS, resolve intra-block suppression serially on the 32x32 diagonal word,
// then apply the kept rows' masks to all later words in parallel.
// ---------------------------------------------------------------------------
__global__ __launch_bounds__(256) void nms_kernel(const u32* __restrict__ mask,
                                                  u32* __restrict__ keepw, int M) {
  __shared__ u32 K[256];          // keep bit-words (M/32 <= 256)
  __shared__ u32 rowm[32 * 256];  // 32 staged mask rows (32 KB)
  const int tid = threadIdx.x;
  const int nw = M >> 5;
  for (int w = tid; w < nw; w += 256) K[w] = 0xFFFFFFFFu;
  __syncthreads();

  for (int b = 0; b < nw; ++b) {
    for (int idx = tid; idx < 32 * nw; idx += 256)
      rowm[idx] = mask[(size_t)(b * 32) * nw + idx];  // rows are contiguous
    __syncthreads();

    // Intra-block serial resolve (computed redundantly by every thread).
    u32 local = K[b];
#pragma unroll 1
    for (int l = 0; l < 32; ++l) {
      if ((local >> l) & 1u) {
        u32 above = (l == 31) ? 0u : (0xFFFFFFFFu << (l + 1));  // only j > i
        local &= ~(rowm[l * nw + b] & above);
      }
    }
    if (tid == 0) K[b] = local;

    // Apply kept rows of this block to all later column words.
    for (int w = b + 1 + tid; w < nw; w += 256) {
      u32 acc = 0u;
#pragma unroll 1
      for (int l = 0; l < 32; ++l)
        if ((local >> l) & 1u) acc |= rowm[l * nw + w];
      K[w] &= ~acc;
    }
    __syncthreads();
  }
  for (int w = tid; w < nw; w += 256) keepw[w] = K[w];
}

// ---------------------------------------------------------------------------
// Kernel 5: per-pair fusion (exact reference formulas, incl. divisions).
// ---------------------------------------------------------------------------
__device__ inline void aabb7(const float* b, float& x1, float& x2, float& z1, float& z2) {
  float dx = b[3] * 0.5f, dz = b[4] * 0.5f;
  float co = cosf(b[6]), si = sinf(b[6]);
  float ex = fabsf(dx * co) + fabsf(dz * si);
  float ez = fabsf(dx * si) + fabsf(dz * co);
  x1 = b[0] - ex; x2 = b[0] + ex; z1 = b[2] - ez; z2 = b[2] + ez;
}

__device__ inline float iou3d(const float* A, const float* B) {
  float ha = A[5], hb = B[5];
  float yov = fminf(A[1] + ha * 0.5f, B[1] + hb * 0.5f) -
              fmaxf(A[1] - ha * 0.5f, B[1] - hb * 0.5f);
  yov = fmaxf(yov, 0.0f);
  float hr = yov / (ha + hb - yov + EPSF);
  float ax1, ax2, az1, az2, bx1, bx2, bz1, bz2;
  aabb7(A, ax1, ax2, az1, az2);
  aabb7(B, bx1, bx2, bz1, bz2);
  float xi = fmaxf(fminf(ax2, bx2) - fmaxf(ax1, bx1), 0.0f);
  float zi = fmaxf(fminf(az2, bz2) - fmaxf(az1, bz1), 0.0f);
  float inter = xi * zi;
  float ua = (ax2 - ax1) * (az2 - az1), ub = (bx2 - bx1) * (bz2 - bz1);
  return inter / (ua + ub - inter + EPSF) * hr;
}

__global__ void fuse_kernel(const float* __restrict__ b1, const float* __restrict__ sboxes,
                            const int* __restrict__ order, const u32* __restrict__ keepw,
                            int N, int M, float* __restrict__ out) {
  int p = blockIdx.x * blockDim.x + threadIdx.x;
  if (p >= M) return;
  int idx = order[p];
  bool inv0 = idx < N;
  const float* A = sboxes + (size_t)p * 8;
  float a[7], fb[7], fused[7];
#pragma unroll
  for (int c = 0; c < 7; ++c) a[c] = A[c];
  if (inv0) {
    const float* B = b1 + (size_t)idx * 7;  // partner = all_boxes[idx + N]
#pragma unroll
    for (int c = 0; c < 7; ++c) fb[c] = B[c];
    float iaa = iou3d(a, a), ibb = iou3d(fb, fb), iab = iou3d(a, fb);  // symmetric
    float ma = fmaxf(iaa, iab), mb = fmaxf(ibb, iab);
    bool ka = ma > THRF, kb = mb > THRF;
    float wa = ka ? ma : 0.0f, wb = kb ? mb : 0.0f;
    float den = wa + wb;
    float dens = (den == 0.0f) ? 1.0f : den;
#pragma unroll
    for (int c = 0; c < 7; ++c) {
      float fw = (a[c] * wa + fb[c] * wb) / dens;
      float fm = (a[c] + fb[c]) * 0.5f;
      fused[c] = (ka || kb) ? fw : fm;
    }
  } else {
#pragma unroll
    for (int c = 0; c < 7; ++c) fused[c] = a[c];
  }
  u32 kbit = (keepw[p >> 5] >> (p & 31)) & 1u;
#pragma unroll
  for (int c = 0; c < 7; ++c) out[(size_t)p * 7 + c] = kbit ? fused[c] : 0.0f;
  out[(size_t)M * 7 + p] = kbit ? 1.0f : 0.0f;  // keep flags after fused boxes
}

// ---------------------------------------------------------------------------
extern "C" void kernel_launch(void* const* d_in, const int* in_sizes, int n_in,
                              void* d_out, int out_size, void* d_ws, size_t ws_size,
                              hipStream_t stream) {
  const float* boxes0 = (const float*)d_in[0];
  const float* boxes1 = (const float*)d_in[1];
  const float* scores0 = (const float*)d_in[2];
  const float* scores1 = (const float*)d_in[3];
  const int N = in_sizes[2];  // 4096
  const int M = 2 * N;        // 8192 (power of two)
  const int nw = M >> 5;      // 256

  char* ws = (char*)d_ws;
  int* order = (int*)ws;                                       // M * 4 B
  float* sboxes = (float*)(ws + (size_t)M * 4);                // M * 32 B
  float* prep = (float*)(ws + (size_t)M * 4 + (size_t)M * 32); // M * 32 B (SoA)
  u32* mask = (u32*)(ws + (size_t)M * 4 + (size_t)M * 64);     // M * nw * 4 B (8 MB)
  u32* keepw = (u32*)((char*)mask + (size_t)M * nw * 4);       // nw * 4 B

  sort_kernel<<<1, 1024, 0, stream>>>(scores0, scores1, N, M, order);
  prep_kernel<<<(M + 255) / 256, 256, 0, stream>>>(boxes0, boxes1, order, N, M, prep, sboxes);

  size_t ldsBytes = (size_t)8 * M * 4;  // 256 KB of the WGP's 320 KB LDS
  (void)hipFuncSetAttribute(reinterpret_cast<const void*>(iou_mask_kernel),
                            hipFuncAttributeMaxDynamicSharedMemorySize, (int)ldsBytes);
  iou_mask_kernel<<<M / 64, 256, ldsBytes, stream>>>(prep, mask, M);

  nms_kernel<<<1, 256, 0, stream>>>(mask, keepw, M);
  fuse_kernel<<<(M + 255) / 256, 256, 0, stream>>>(boxes1, sboxes, order, keepw, N, M,
                                                   (float*)d_out);
}